// Nbp_16750372454696
// MI455X (gfx1250) — compile-verified
//
#include <hip/hip_runtime.h>
#include <math.h>

typedef __attribute__((ext_vector_type(2))) float v2f;
typedef __attribute__((ext_vector_type(8))) float v8f;

constexpr int Mdim = 512;
constexpr int Ndim = 2048;
constexpr int Bdim = 4;
constexpr int Ldim = 20;
constexpr float EPSC = 1e-6f;

// ---------------------------------------------------------------------------
// Weighted column-sum over M via WMMA "ones" trick.
//   out[b][n] = sum_m src[b][m][n] * w[m][n]   (+ llrs[n]*w_llr[n] if given)
// One wave per (b, 16-column tile). A = ones(16x4) so every row of C
// accumulates the column sums of the 4x16 product tile fed as B.
// B (4x16 f32, 2 VGPRs, wave32) layout (C-layout analog):
//   v0: lanes 0-15 -> K=0, lanes 16-31 -> K=2 ; v1: K=1 / K=3.
// C (16x16 f32): VGPR0 lanes 0-15 hold row M=0, N=lane.
// ---------------------------------------------------------------------------
__global__ void colsum_wmma_kernel(const float* __restrict__ src,
                                   const float* __restrict__ w,
                                   const float* __restrict__ llrs,    // nullable
                                   const float* __restrict__ w_llr,   // nullable
                                   float* __restrict__ out) {
  const int wave   = (blockIdx.x * blockDim.x + threadIdx.x) >> 5;
  const int lane   = threadIdx.x & 31;
  const int ntiles = Ndim / 16;
  const int b      = wave / ntiles;
  const int nt     = wave % ntiles;
  const int col    = nt * 16 + (lane & 15);
  const int krow   = (lane >> 4) * 2;          // {0,1} vs {2,3} row pair

  const float* mbase = src + ((size_t)b * Mdim) * Ndim + col;
  const float* wbase = w + col;

  v8f c = {};
  v2f ones; ones.x = 1.0f; ones.y = 1.0f;

  for (int m = 0; m < Mdim; m += 4) {
    const float* mp = mbase + (size_t)(m + krow) * Ndim;
    const float* wp = wbase + (size_t)(m + krow) * Ndim;
    __builtin_prefetch(wp + 8 * Ndim, 0, 1);   // stream-ahead on weights
    v2f bv;
    bv.x = mp[0]    * wp[0];                    // row m+krow
    bv.y = mp[Ndim] * wp[Ndim];                 // row m+krow+1
    c = __builtin_amdgcn_wmma_f32_16x16x4_f32(false, ones, false, bv,
                                              (short)0, c, false, false);
  }

  if (lane < 16) {
    float v = c[0];                             // C row M=0 -> column sums
    if (llrs) v += llrs[col] * w_llr[col];
    out[(size_t)b * Ndim + col] = v;
  }
}

// ---------------------------------------------------------------------------
// Row products of d = clip(tanh(0.5*msgs_en)). One wave per (b,m) row.
// ---------------------------------------------------------------------------
__global__ void rowprod_kernel(const float* __restrict__ src,
                               const float* __restrict__ colsum,
                               const float* __restrict__ llrs,
                               const float* __restrict__ w_llr,
                               const float* __restrict__ Hmat,
                               float* __restrict__ rowprod) {
  const int wave = (blockIdx.x * blockDim.x + threadIdx.x) >> 5;
  const int lane = threadIdx.x & 31;
  const int b = wave / Mdim;
  const int m = wave % Mdim;

  const float* mrow = src + ((size_t)b * Mdim + m) * Ndim;
  const float* hrow = Hmat + (size_t)m * Ndim;
  const float* cs   = colsum + (size_t)b * Ndim;

  float prod = 1.0f;
  for (int n = lane; n < Ndim; n += 32) {
    float en = (llrs[n] * w_llr[n] + cs[n]) * hrow[n] - mrow[n];
    float d = tanhf(0.5f * en);
    d = (d == 0.0f) ? 1.0f : d;
    d = fminf(fmaxf(d, -1.0f + EPSC), 1.0f - EPSC);
    prod *= d;
  }
#pragma unroll
  for (int off = 16; off > 0; off >>= 1)
    prod *= __shfl_xor(prod, off, 32);
  if (lane == 0) rowprod[(size_t)b * Mdim + m] = prod;
}

// ---------------------------------------------------------------------------
// msgs_new = 2*arctanh(rowprod/d) * sign * H + rw * msgs_dn   (elementwise)
// ---------------------------------------------------------------------------
__global__ void update_kernel(const float* __restrict__ src,
                              const float* __restrict__ colsum,
                              const float* __restrict__ rowprod,
                              const float* __restrict__ llrs,
                              const float* __restrict__ w_llr,
                              const float* __restrict__ Hmat,
                              const int*   __restrict__ syndromes,
                              const float* __restrict__ resid_w, int layer,
                              float* __restrict__ dst) {
  const size_t idx = (size_t)blockIdx.x * blockDim.x + threadIdx.x;
  const int n  = (int)(idx % Ndim);
  const int bm = (int)(idx / Ndim);           // b*M + m
  const int m  = bm % Mdim;
  const int b  = bm / Mdim;

  const float rw  = resid_w[layer];
  const float h   = Hmat[(size_t)m * Ndim + n];
  const float mdn = src[idx];

  float en = (llrs[n] * w_llr[n] + colsum[(size_t)b * Ndim + n]) * h - mdn;
  float d = tanhf(0.5f * en);
  d = (d == 0.0f) ? 1.0f : d;
  d = fminf(fmaxf(d, -1.0f + EPSC), 1.0f - EPSC);

  float r    = rowprod[bm] / d;
  float sgn  = 1.0f - 2.0f * (float)syndromes[bm];
  float atn2 = logf((1.0f + r) / (1.0f - r));   // 2*arctanh(r)
  dst[idx] = atn2 * sgn * h + rw * mdn;
}

// ---------------------------------------------------------------------------
// Per-layer loss: sum over (b,n) of softplus(x) - (1-err)*x. Single block.
// ---------------------------------------------------------------------------
__global__ void loss_layer_kernel(const float* __restrict__ beliefs,
                                  const int*   __restrict__ errors,
                                  float* __restrict__ loss_raw, int layer) {
  __shared__ float sm[256];
  float acc = 0.0f;
  for (int i = threadIdx.x; i < Bdim * Ndim; i += 256) {
    float x = beliefs[i];
    float e = (float)errors[i];
    float sp = fmaxf(x, 0.0f) + log1pf(expf(-fabsf(x)));   // stable softplus
    acc += sp - (1.0f - e) * x;
  }
  sm[threadIdx.x] = acc;
  __syncthreads();
  for (int st = 128; st > 0; st >>= 1) {
    if (threadIdx.x < st) sm[threadIdx.x] += sm[threadIdx.x + st];
    __syncthreads();
  }
  if (threadIdx.x == 0) loss_raw[layer] = sm[0];
}

// ---------------------------------------------------------------------------
// out = (sum_l softmax(rhos)[l] * loss_raw[l]) / B
// ---------------------------------------------------------------------------
__global__ void finalize_kernel(const float* __restrict__ rhos,
                                const float* __restrict__ loss_raw,
                                float* __restrict__ out) {
  if (threadIdx.x == 0) {
    float mx = rhos[0];
    for (int l = 1; l < Ldim; ++l) mx = fmaxf(mx, rhos[l]);
    float se = 0.0f;
    for (int l = 0; l < Ldim; ++l) se += expf(rhos[l] - mx);
    float total = 0.0f;
    for (int l = 0; l < Ldim; ++l)
      total += (expf(rhos[l] - mx) / se) * loss_raw[l];
    out[0] = total / (float)Bdim;
  }
}

extern "C" void kernel_launch(void* const* d_in, const int* in_sizes, int n_in,
                              void* d_out, int out_size, void* d_ws, size_t ws_size,
                              hipStream_t stream) {
  const int*   syndromes  = (const int*)  d_in[0];  // (B,M,1)
  const int*   errors     = (const int*)  d_in[1];  // (B,N)
  const float* Hmat       = (const float*)d_in[2];  // (M,N)
  const float* llrs       = (const float*)d_in[3];  // (N,)
  const float* w_de_all   = (const float*)d_in[4];  // (L,M,N)
  const float* w_llr_all  = (const float*)d_in[5];  // (L,N)
  const float* mw_de_all  = (const float*)d_in[6];  // (L,M,N)
  const float* mw_llr_all = (const float*)d_in[7];  // (L,N)
  const float* rhos       = (const float*)d_in[8];  // (L,)
  const float* resid_w    = (const float*)d_in[9];  // (L,)
  (void)in_sizes; (void)n_in; (void)out_size; (void)ws_size;

  const size_t msz = (size_t)Bdim * Mdim * Ndim;
  float* msgsA    = (float*)d_ws;
  float* msgsB    = msgsA + msz;
  float* colsum   = msgsB + msz;
  float* rowprod  = colsum + (size_t)Bdim * Ndim;
  float* beliefs  = rowprod + (size_t)Bdim * Mdim;
  float* loss_raw = beliefs + (size_t)Bdim * Ndim;

  hipMemsetAsync(msgsA, 0, msz * sizeof(float), stream);  // init messages = 0

  const int csBlocks = (Bdim * (Ndim / 16)) / 8;  // 8 waves per 256-thread block
  const int rpBlocks = (Bdim * Mdim) / 8;
  const int upBlocks = (int)(msz / 256);

  float* src = msgsA;
  float* dst = msgsB;
  for (int l = 0; l < Ldim; ++l) {
    const float* w_de   = w_de_all   + (size_t)l * Mdim * Ndim;
    const float* w_llr  = w_llr_all  + (size_t)l * Ndim;
    const float* mw_de  = mw_de_all  + (size_t)l * Mdim * Ndim;
    const float* mw_llr = mw_llr_all + (size_t)l * Ndim;

    colsum_wmma_kernel<<<csBlocks, 256, 0, stream>>>(src, w_de, nullptr, nullptr,
                                                     colsum);
    rowprod_kernel<<<rpBlocks, 256, 0, stream>>>(src, colsum, llrs, w_llr, Hmat,
                                                 rowprod);
    update_kernel<<<upBlocks, 256, 0, stream>>>(src, colsum, rowprod, llrs, w_llr,
                                                Hmat, syndromes, resid_w, l, dst);
    colsum_wmma_kernel<<<csBlocks, 256, 0, stream>>>(dst, mw_de, llrs, mw_llr,
                                                     beliefs);
    loss_layer_kernel<<<1, 256, 0, stream>>>(beliefs, errors, loss_raw, l);

    float* t = src; src = dst; dst = t;
  }

  finalize_kernel<<<1, 32, 0, stream>>>(rhos, loss_raw, (float*)d_out);
}